// CTCLayer_50895362457745
// MI455X (gfx1250) — compile-verified
//
#include <hip/hip_runtime.h>
#include <hip/hip_bf16.h>

// Problem constants (match reference)
#define B_    256
#define T_    512
#define L_    48
#define C_    512
#define S_    97
#define BLANK_ 511
#define NEGV  (-1.0e30f)
#define EPS_  (1e-7f)
#define LN2_  0.6931471805599453f

// Async prefetch pipeline: 7-step lookahead, 8-slot LDS ring
#define LOOK  7
#define RING  8
#define WAITN 18   // 3*(LOOK-1): oldest 3 loads guaranteed complete

// cpol: TH_NT (non-temporal) — y_pred is 256MB streamed once, > 192MB L2
#define CPOL_NT 1

#if defined(__HIP_DEVICE_COMPILE__) && __has_builtin(__builtin_amdgcn_global_load_async_to_lds_b32)
#define HAS_ASYNC 1
#else
#define HAS_ASYNC 0
#endif

typedef __attribute__((address_space(1))) int* as1_i32p;   // global int*
typedef __attribute__((address_space(3))) int* as3_i32p;   // LDS int*

__device__ __forceinline__ void async_gather1(const float* g, float* l) {
#if HAS_ASYNC
  __builtin_amdgcn_global_load_async_to_lds_b32(
      (as1_i32p)(unsigned long long)g, (as3_i32p)l, 0, CPOL_NT);
#else
  *l = __builtin_nontemporal_load(g);   // sync fallback (still correct)
#endif
}

__device__ __forceinline__ void wait_async() {
#if HAS_ASYNC
#if __has_builtin(__builtin_amdgcn_s_wait_asynccnt)
  __builtin_amdgcn_s_wait_asynccnt(WAITN);
#else
  asm volatile("s_wait_asynccnt " "18" ::: "memory");
#endif
#endif
}

// base-2 log-sum-exp (v_exp_f32 = 2^x, v_log_f32 = log2(x))
__device__ __forceinline__ float xexp2(float x) { return __builtin_amdgcn_exp2f(x); }
__device__ __forceinline__ float xlog2(float x) { return __builtin_amdgcn_logf(x); }

__device__ __forceinline__ float lse2(float a, float b) {
  float m = fmaxf(a, b);
  return m + xlog2(xexp2(a - m) + xexp2(b - m));
}
__device__ __forceinline__ float lse3(float a, float b, float c) {
  float m = fmaxf(fmaxf(a, b), c);
  return m + xlog2(xexp2(a - m) + xexp2(b - m) + xexp2(c - m));
}

__global__ __launch_bounds__(32)
void ctc_loss_kernel(const int* __restrict__ y_true,
                     const float* __restrict__ y_pred,
                     float* __restrict__ out) {
  __shared__ float ring[RING][96];          // 3 floats per lane per slot
  const int b    = blockIdx.x;
  const int lane = threadIdx.x;             // wave32: one wave per block

  // ---- per-lane extended-label classes and skip flags --------------------
  // lane owns states s = 4*lane + k, k=0..3.  k even -> blank.
  // k=1 -> label j=2*lane ; k=3 -> label j=2*lane+1
  int  c1 = BLANK_, c3 = BLANK_;
  bool sk1 = false, sk3 = false;
  if (lane <= 23) {                         // states <= 95 carry labels
    const int* lb = y_true + b * L_;
    int j  = 2 * lane;
    int l1 = lb[j];
    int l2 = lb[j + 1];
    c1 = l1; c3 = l2;
    sk3 = (l2 != l1);
    if (lane > 0) { int l0 = lb[j - 1]; sk1 = (l1 != l0); }
    // s==1 (lane 0, k=1): alpha[-1] pad -> sk1 stays false
  }

  const float* base = y_pred + (size_t)b * (T_ * C_);
  const int lo = lane * 3;

  // ---- prologue: issue async gathers for t = 0..LOOK-1 -------------------
  for (int t = 0; t < LOOK; ++t) {
    const float* row = base + (size_t)t * C_;
    async_gather1(row + BLANK_, &ring[t][lo + 0]);
    async_gather1(row + c1,     &ring[t][lo + 1]);
    async_gather1(row + c3,     &ring[t][lo + 2]);
  }

  volatile float* rv = &ring[0][0];

  // ---- t = 0: init alpha --------------------------------------------------
  wait_async();
  {
    float pB = rv[lo + 0];
    float p1 = rv[lo + 1];
    float lpB = xlog2(pB + EPS_);
    float lp1 = xlog2(p1 + EPS_);
    float a0 = (lane == 0) ? lpB : NEGV;   // state 0 (blank)
    float a1 = (lane == 0) ? lp1 : NEGV;   // state 1 (first label)
    float a2 = NEGV, a3 = NEGV;

    // issue t = LOOK
    {
      const float* row = base + (size_t)LOOK * C_;
      float* sl = &ring[LOOK & (RING - 1)][lo];
      async_gather1(row + BLANK_, sl + 0);
      async_gather1(row + c1,     sl + 1);
      async_gather1(row + c3,     sl + 2);
    }

    // ---- main scan: t = 1..T-1 -------------------------------------------
#pragma unroll 2
    for (int t = 1; t < T_; ++t) {
      wait_async();
      const int o = (t & (RING - 1)) * 96 + lo;
      float qB = rv[o + 0];
      float q1 = rv[o + 1];
      float q3 = rv[o + 2];

      // keep issue rate constant so the immediate wait count stays valid;
      // wrapped tail issues land in slots that are never read again.
      {
        int ta = t + LOOK; if (ta >= T_) ta -= T_;
        const float* row = base + (size_t)ta * C_;
        float* sl = &ring[(t + LOOK) & (RING - 1)][lo];
        async_gather1(row + BLANK_, sl + 0);
        async_gather1(row + c1,     sl + 1);
        async_gather1(row + c3,     sl + 2);
      }

      float lB = xlog2(qB + EPS_);
      float l1 = xlog2(q1 + EPS_);
      float l3 = xlog2(q3 + EPS_);

      float pm1 = __shfl_up(a3, 1, 32);    // alpha_old[4*lane - 1]
      if (lane == 0) pm1 = NEGV;

      float n0 = lse2(a0, pm1)                        + lB;  // even: no skip
      float n1 = lse3(a1, a0, sk1 ? pm1 : NEGV)       + l1;
      float n2 = lse2(a2, a1)                         + lB;  // even: no skip
      float n3 = lse3(a3, a2, sk3 ? a1  : NEGV)       + l3;
      a0 = n0; a1 = n1; a2 = n2; a3 = n3;
    }

    // ---- finalize: ll = lse(alpha[96], alpha[95]) ---------------------------
    float v95 = __shfl(a3, 23, 32);        // state 95 = lane23.k3
    float v96 = __shfl(a0, 24, 32);        // state 96 = lane24.k0
    if (lane == 0) {
      out[b] = -(lse2(v95, v96) * LN2_);   // convert log2 -> ln
    }
  }
}

extern "C" void kernel_launch(void* const* d_in, const int* in_sizes, int n_in,
                              void* d_out, int out_size, void* d_ws, size_t ws_size,
                              hipStream_t stream) {
  (void)in_sizes; (void)n_in; (void)d_ws; (void)ws_size; (void)out_size;
  const int*   y_true = (const int*)d_in[0];
  const float* y_pred = (const float*)d_in[1];
  float*       out    = (float*)d_out;
  ctc_loss_kernel<<<B_, 32, 0, stream>>>(y_true, y_pred, out);
}